// Thing_91302414778504
// MI455X (gfx1250) — compile-verified
//
#include <hip/hip_runtime.h>

typedef __attribute__((ext_vector_type(2))) float v2f;
typedef __attribute__((ext_vector_type(8))) float v8f;

#define NWAVES   8
#define NTHREADS 256

// ---------------- workspace layout (floats, per sample) ----------------
constexpr int OFF_Y0   = 0;        // 4x128x128 pre-act of enc conv0
constexpr int OFF_H0   = 65536;    // 4x64x64  pooled
constexpr int OFF_Y1   = 81920;    // 4x64x64
constexpr int OFF_H1   = 98304;    // 4x32x32
constexpr int OFF_Y2   = 102400;   // 4x32x32
constexpr int OFF_H2   = 106496;   // 4x16x16
constexpr int OFF_Y3   = 107520;   // 4x16x16
constexpr int OFF_H3   = 108544;   // 4x8x8
constexpr int OFF_Y4   = 108800;   // 4x8x8
constexpr int OFF_H4   = 109056;   // 4x4x4
constexpr int OFF_Y5   = 109120;   // 4x4x4
constexpr int OFF_H5   = 109184;   // 4x4x4
constexpr int OFF_ZPRE = 109248;   // 32
constexpr int OFF_Z    = 109280;   // 32
constexpr int OFF_DPRE = 109312;   // 64
constexpr int OFF_DVEC = 109376;   // 64
constexpr int OFF_DY0  = 109440;   // 4x8x8
constexpr int OFF_DH0  = 109696;
constexpr int OFF_DY1  = 109952;   // 4x16x16
constexpr int OFF_DH1  = 110976;
constexpr int OFF_DY2  = 112000;   // 4x32x32
constexpr int OFF_DH2  = 116096;
constexpr int OFF_DY3  = 120192;   // 4x64x64
constexpr int OFF_DH3  = 136576;
constexpr int OFF_DY4  = 152960;   // 4x128x128
constexpr int OFF_DH4  = 218496;
constexpr int OFF_OUT1 = 284032;   // 1x128x128 first-pass output
constexpr int OFF_GA   = 300416;   // grad ping (65536)
constexpr int OFF_GB   = 365952;   // grad pong (65536)
constexpr int OFF_UP   = 431488;   // upsample scratch (65536)
constexpr int OFF_GRAD = 497024;   // packed param grads (5760)
constexpr int OFF_UPD  = 502784;   // packed updated params (5760)
constexpr int OFF_WF   = 508544;   // flipped-weight scratch (160)
constexpr int OFF_STMP = 508704;   // linear-bwd scratch (64)
constexpr size_t PER_SAMPLE = 524288;  // 2 MB / sample

// packed param offsets
constexpr int P_ENCW0 = 0;      // 36
constexpr int P_ENCW  = 36;     // 720
constexpr int P_ENCB  = 756;    // 24
constexpr int P_ELW   = 780;    // 2048
constexpr int P_ELB   = 2828;   // 32
constexpr int P_DLW   = 2860;   // 2048
constexpr int P_DLB   = 4908;   // 64
constexpr int P_DECW  = 4972;   // 720
constexpr int P_DECB  = 5692;   // 20
constexpr int P_DWO   = 5712;   // 36
constexpr int P_DBO   = 5748;   // 1
constexpr int P_TOTAL = 5749;

constexpr int ENCY[6] = {OFF_Y0, OFF_Y1, OFF_Y2, OFF_Y3, OFF_Y4, OFF_Y5};
constexpr int ENCH[6] = {OFF_H0, OFF_H1, OFF_H2, OFF_H3, OFF_H4, OFF_H5};
constexpr int DECY[5] = {OFF_DY0, OFF_DY1, OFF_DY2, OFF_DY3, OFF_DY4};
constexpr int DECH[5] = {OFF_DH0, OFF_DH1, OFF_DH2, OFF_DH3, OFF_DH4};

struct Pm {
  const float *ew0, *ew, *eb, *elw, *elb, *dlw, *dlb, *dw, *db, *dwo, *dbo;
};

__device__ __forceinline__ float sigm_(float x) { return 1.0f / (1.0f + __expf(-x)); }
__device__ __forceinline__ float swish_(float x) { return x * sigm_(x); }
__device__ __forceinline__ float dswish_(float x) {
  float s = sigm_(x);
  return s * (1.0f + x * (1.0f - s));
}

// Branchless im2col gather: clamp address into range, load unconditionally,
// select 0 with v_cndmask. k = ci*9 + kh*3 + kw over square HxH map (wl=log2 H).
template <int K>
__device__ __forceinline__ float colval(const float* __restrict__ X, int pos, int k,
                                        int H, int wl) {
  const int kk = (k < K) ? k : 0;
  int ci = kk / 9;
  int r  = kk - ci * 9;
  int kh = r / 3;
  int kw = r - kh * 3;
  int y = (pos >> wl) + kh - 1;
  int x = (pos & (H - 1)) + kw - 1;
  bool inb = ((unsigned)y < (unsigned)H) & ((unsigned)x < (unsigned)H) & (k < K);
  int yc = y < 0 ? 0 : (y >= H ? H - 1 : y);
  int xc = x < 0 ? 0 : (x >= H ? H - 1 : x);
  float v = X[ci * (H << wl) + (yc << wl) + xc];
  return inb ? v : 0.0f;
}

// ---- 3x3 SAME conv, implicit GEMM on V_WMMA_F32_16X16X4_F32 ----
// M = 16 spatial positions per wave-tile, N = out channels, K = CIN*9 (unrolled).
// Weights (B operand, tile-invariant) preloaded into registers once.
template <int CIN, int COUT>
__device__ __forceinline__ void conv3x3_fwd(const float* __restrict__ X,
                                            float* __restrict__ Y,
                                            const float* __restrict__ Wt,
                                            const float* __restrict__ Bs, int H,
                                            int wl) {
  constexpr int K      = CIN * 9;
  constexpr int KSTEPS = (K + 3) / 4;
  const int HW    = H << wl;
  const int lane  = threadIdx.x & 31;
  const int wave  = threadIdx.x >> 5;
  const int r15   = lane & 15;
  const int half  = lane >> 4;
  const int half2 = half << 1;
  const int nt    = HW >> 4;
  const int rc    = (r15 < COUT) ? r15 : 0;

  v2f breg[KSTEPS];
#pragma unroll
  for (int s = 0; s < KSTEPS; ++s) {
    const int kb = s * 4 + half2;
    float w0 = Wt[rc * K + ((kb < K) ? kb : 0)];
    float w1 = Wt[rc * K + ((kb + 1 < K) ? kb + 1 : 0)];
    breg[s].x = ((kb < K) & (r15 < COUT)) ? w0 : 0.0f;
    breg[s].y = ((kb + 1 < K) & (r15 < COUT)) ? w1 : 0.0f;
  }
  const float bs = (r15 < COUT && Bs) ? Bs[r15] : 0.0f;

  for (int tile = wave; tile < nt; tile += NWAVES) {
    const int pos = tile * 16 + r15;  // A row (M) = spatial position
    __builtin_prefetch(&X[(tile + NWAVES) * 16], 0, 1);
    v8f acc = {};
#pragma unroll
    for (int s = 0; s < KSTEPS; ++s) {
      v2f a;
      a.x = colval<K>(X, pos, s * 4 + half2, H, wl);
      a.y = colval<K>(X, pos, s * 4 + half2 + 1, H, wl);
      acc = __builtin_amdgcn_wmma_f32_16x16x4_f32(false, a, false, breg[s], (short)0,
                                                  acc, false, false);
    }
    if (r15 < COUT) {  // D: N(channel) across lanes, M(position) in regs
      float4 lo = make_float4(acc[0] + bs, acc[1] + bs, acc[2] + bs, acc[3] + bs);
      float4 hi = make_float4(acc[4] + bs, acc[5] + bs, acc[6] + bs, acc[7] + bs);
      float* dst = Y + r15 * HW + tile * 16 + (half << 3);
      *reinterpret_cast<float4*>(dst)     = lo;
      *reinterpret_cast<float4*>(dst + 4) = hi;
    }
  }
}

// weight grad: gW[o][k] += sum_pos dY[o][pos] * im2col(pos,k)
// WMMA with GEMM-K = spatial positions; per-wave partials merged via atomics.
template <int CIN, int COUT>
__device__ __forceinline__ void conv3x3_dw(const float* __restrict__ dY,
                                           const float* __restrict__ X, int H, int wl,
                                           float* __restrict__ gW) {
  constexpr int K   = CIN * 9;
  constexpr int NCT = (K + 15) / 16;
  const int HW   = H << wl;
  const int lane = threadIdx.x & 31;
  const int wave = threadIdx.x >> 5;
  const int r15  = lane & 15;
  const int half = lane >> 4;
  const int rc   = (r15 < COUT) ? r15 : 0;
#pragma unroll
  for (int ct = 0; ct < NCT; ++ct) {
    const int kc = ct * 16 + r15;  // B column = weight-k index
    v8f acc = {};
    for (int p0 = wave * 4; p0 < HW; p0 += 4 * NWAVES) {
      const int pb = p0 + (half << 1);
      v2f a, b;
      float d0 = dY[rc * HW + pb];
      float d1 = dY[rc * HW + pb + 1];
      a.x = (r15 < COUT) ? d0 : 0.0f;
      a.y = (r15 < COUT) ? d1 : 0.0f;
      b.x = colval<K>(X, pb, kc, H, wl);
      b.y = colval<K>(X, pb + 1, kc, H, wl);
      acc = __builtin_amdgcn_wmma_f32_16x16x4_f32(false, a, false, b, (short)0, acc,
                                                  false, false);
    }
    if (half == 0 && kc < K) {
#pragma unroll
      for (int g = 0; g < 8; ++g)
        if (g < COUT) atomicAdd(&gW[g * K + kc], acc[g]);
    }
  }
}

__device__ void conv3x3_db(const float* __restrict__ dY, int Cout, int HW,
                           float* __restrict__ gB) {
  for (int c = 0; c < Cout; ++c) {
    float s = 0.0f;
    for (int i = threadIdx.x; i < HW; i += NTHREADS) s += dY[c * HW + i];
#pragma unroll
    for (int o = 16; o > 0; o >>= 1) s += __shfl_xor(s, o, 32);
    if ((threadIdx.x & 31) == 0) atomicAdd(&gB[c], s);
  }
}

// input grad: dX = conv_SAME(dY, flip+transpose(W)); flipped weights staged in
// per-sample global scratch (keeps every pointer in the global address space).
template <int CIN, int COUT>
__device__ __forceinline__ void conv3x3_dx(const float* __restrict__ dY,
                                           const float* __restrict__ Wt, int H, int wl,
                                           float* __restrict__ dX, float* wf) {
  __syncthreads();
  for (int i = threadIdx.x; i < CIN * COUT * 9; i += NTHREADS) {
    int ci = i / (COUT * 9);
    int r  = i - ci * (COUT * 9);
    int o  = r / 9;
    int rr = r - o * 9;
    int kh = rr / 3, kw = rr - kh * 3;
    wf[i] = Wt[(o * CIN + ci) * 9 + (2 - kh) * 3 + (2 - kw)];
  }
  __syncthreads();
  conv3x3_fwd<COUT, CIN>(dY, dX, wf, nullptr, H, wl);
  __syncthreads();
}

// ---------------- elementwise helpers (all shift/mask index math) ----------------
__device__ void k_swish(const float* Y, float* O, int n) {
  for (int i = threadIdx.x; i < n; i += NTHREADS) O[i] = swish_(Y[i]);
}

// O (C x h x h) = maxpool2x2(swish(Y (C x H x H))), H = 1<<wl
__device__ void k_pool_swish(const float* Y, float* O, int C, int H, int wl) {
  const int h   = H >> 1;
  const int wlh = wl - 1;
  const int n   = C << (2 * wlh);
  for (int i = threadIdx.x; i < n; i += NTHREADS) {
    int c = i >> (2 * wlh);
    int r = i & ((1 << (2 * wlh)) - 1);
    int y = r >> wlh, x = r & (h - 1);
    const float* p = Y + (c << (2 * wl)) + ((y << 1) << wl) + (x << 1);
    float v0 = swish_(p[0]), v1 = swish_(p[1]);
    float v2 = swish_(p[H]), v3 = swish_(p[H + 1]);
    O[i] = fmaxf(fmaxf(v0, v1), fmaxf(v2, v3));
  }
}

// grad through pool(swish(.)): route to first argmax in window, then * swish'
__device__ void k_pool_bwd(const float* gP, const float* Y, float* gO, int C, int H,
                           int wl) {
  const int wlh = wl - 1;
  const int n   = C << (2 * wl);
  for (int i = threadIdx.x; i < n; i += NTHREADS) {
    int c  = i >> (2 * wl);
    int r  = i & ((1 << (2 * wl)) - 1);
    int yy = r >> wl, xx = r & (H - 1);
    int y = yy >> 1, x = xx >> 1;
    const float* yb = Y + (c << (2 * wl)) + ((y << 1) << wl) + (x << 1);
    float v0 = swish_(yb[0]), v1 = swish_(yb[1]);
    float v2 = swish_(yb[H]), v3 = swish_(yb[H + 1]);
    float m  = fmaxf(fmaxf(v0, v1), fmaxf(v2, v3));
    int jstar = (v0 == m) ? 0 : (v1 == m) ? 1 : (v2 == m) ? 2 : 3;
    int jme   = ((yy & 1) << 1) | (xx & 1);
    float g = (jme == jstar) ? gP[(c << (2 * wlh)) + (y << wlh) + x] : 0.0f;
    gO[i] = g * dswish_(Y[i]);
  }
}

// U (C x 2h x 2h) = nearest-up2(Hin (C x h x h)), h = 1<<wlh
__device__ void k_up(const float* Hin, float* U, int C, int h, int wlh) {
  const int wlH = wlh + 1;
  const int H   = h << 1;
  const int n   = C << (2 * wlH);
  for (int i = threadIdx.x; i < n; i += NTHREADS) {
    int c = i >> (2 * wlH);
    int r = i & ((1 << (2 * wlH)) - 1);
    int y = r >> wlH, x = r & (H - 1);
    U[i] = Hin[(c << (2 * wlh)) + ((y >> 1) << wlh) + (x >> 1)];
  }
}

__device__ void k_up_bwd(const float* gU, float* gH, int C, int h, int wlh) {
  const int wlH = wlh + 1;
  const int H   = h << 1;
  const int n   = C << (2 * wlh);
  for (int i = threadIdx.x; i < n; i += NTHREADS) {
    int c = i >> (2 * wlh);
    int r = i & ((1 << (2 * wlh)) - 1);
    int y = r >> wlh, x = r & (h - 1);
    const float* p = gU + (c << (2 * wlH)) + ((y << 1) << wlH) + (x << 1);
    gH[i] = p[0] + p[1] + p[H] + p[H + 1];
  }
}

__device__ void lin_fwd(const float* Wm, const float* b, const float* xin, float* ypre,
                        int M, int Kd) {
  for (int i = threadIdx.x; i < M; i += NTHREADS) {
    float s = b[i];
    for (int j = 0; j < Kd; ++j) s += Wm[i * Kd + j] * xin[j];
    ypre[i] = s;
  }
}

// backward through swish(W x + b): accumulates gW,gB, writes gx (may alias gio)
__device__ void lin_bwd(const float* Wm, const float* ypre, const float* xin,
                        float* gio, float* gW, float* gB, float* gx, int M, int Kd,
                        float* sTmp) {
  for (int i = threadIdx.x; i < M; i += NTHREADS) {
    float gp = gio[i] * dswish_(ypre[i]);
    sTmp[i]  = gp;
    gB[i]   += gp;
  }
  __syncthreads();
  for (int t = threadIdx.x; t < M * Kd; t += NTHREADS) {
    int i2 = t / Kd, j = t - i2 * Kd;
    gW[t] += sTmp[i2] * xin[j];
  }
  for (int j = threadIdx.x; j < Kd; j += NTHREADS) {
    float acc = 0.0f;
    for (int i2 = 0; i2 < M; ++i2) acc += Wm[i2 * Kd + j] * sTmp[i2];
    gx[j] = acc;
  }
  __syncthreads();
}

// ---------------- autoencoder forward (stashes everything) ----------------
__device__ void ae_fwd(const float* xs, const Pm& p, float* S, float* outp) {
  conv3x3_fwd<1, 4>(xs, S + OFF_Y0, p.ew0, p.eb, 128, 7);
  __syncthreads();
  k_pool_swish(S + OFF_Y0, S + OFF_H0, 4, 128, 7);
  __syncthreads();
  int off_in = OFF_H0, hsz = 64, wl = 6;
  for (int i = 0; i < 5; ++i) {
    conv3x3_fwd<4, 4>(S + off_in, S + ENCY[i + 1], p.ew + i * 144, p.eb + (i + 1) * 4,
                      hsz, wl);
    __syncthreads();
    if (i < 4) {
      k_pool_swish(S + ENCY[i + 1], S + ENCH[i + 1], 4, hsz, wl);
      hsz >>= 1;
      wl -= 1;
    } else {
      k_swish(S + ENCY[i + 1], S + ENCH[i + 1], 4 * hsz * hsz);
    }
    off_in = ENCH[i + 1];
    __syncthreads();
  }
  lin_fwd(p.elw, p.elb, S + OFF_H5, S + OFF_ZPRE, 32, 64);
  __syncthreads();
  k_swish(S + OFF_ZPRE, S + OFF_Z, 32);
  __syncthreads();
  lin_fwd(p.dlw, p.dlb, S + OFF_Z, S + OFF_DPRE, 64, 32);
  __syncthreads();
  k_swish(S + OFF_DPRE, S + OFF_DVEC, 64);
  __syncthreads();
  int cur = OFF_DVEC, cs = 4, cwl = 2;
  for (int j = 0; j < 5; ++j) {
    k_up(S + cur, S + OFF_UP, 4, cs, cwl);
    __syncthreads();
    const int os = cs << 1, owl = cwl + 1;
    conv3x3_fwd<4, 4>(S + OFF_UP, S + DECY[j], p.dw + j * 144, p.db + j * 4, os, owl);
    __syncthreads();
    k_swish(S + DECY[j], S + DECH[j], 4 * os * os);
    __syncthreads();
    cur = DECH[j];
    cs  = os;
    cwl = owl;
  }
  conv3x3_fwd<4, 1>(S + OFF_DH4, outp, p.dwo, p.dbo, 128, 7);
  __syncthreads();
}

// ---------------- backward: param grads into G ----------------
__device__ void ae_bwd(const float* xs, const Pm& p, float* S, float* G) {
  float* g0 = S + OFF_GB;
  float* g1 = S + OFF_GA;
  float* wf = S + OFF_WF;
  float* st = S + OFF_STMP;
  const float scale = 2.0f / (float)(128 * 16384);
  for (int i = threadIdx.x; i < 16384; i += NTHREADS)
    g1[i] = scale * (S[OFF_OUT1 + i] - xs[i]);
  __syncthreads();
  // final conv (dec_w_out): Cin=4, Cout=1 @128
  conv3x3_db(g1, 1, 16384, G + P_DBO);
  conv3x3_dw<4, 1>(g1, S + OFF_DH4, 128, 7, G + P_DWO);
  __syncthreads();
  conv3x3_dx<4, 1>(g1, p.dwo, 128, 7, g0, wf);  // g0 = d(DH4)
  // decoder, j = 4..0
  int cs = 128, cwl = 7;
  for (int j = 4; j >= 0; --j) {
    const int n = 4 * cs * cs;
    for (int i = threadIdx.x; i < n; i += NTHREADS)
      g1[i] = g0[i] * dswish_(S[DECY[j] + i]);
    __syncthreads();
    const int hs = cs >> 1, hwl = cwl - 1;
    k_up(S + ((j > 0) ? DECH[j - 1] : OFF_DVEC), S + OFF_UP, 4, hs, hwl);
    __syncthreads();
    conv3x3_db(g1, 4, cs * cs, G + P_DECB + j * 4);
    conv3x3_dw<4, 4>(g1, S + OFF_UP, cs, cwl, G + P_DECW + j * 144);
    __syncthreads();
    conv3x3_dx<4, 4>(g1, p.dw + j * 144, cs, cwl, g0, wf);  // g0 = d(upsampled)
    k_up_bwd(g0, g1, 4, hs, hwl);
    __syncthreads();
    float* t = g0; g0 = g1; g1 = t;  // g0 = d(prev post-act)
    cs  = hs;
    cwl = hwl;
  }
  // linear layers (g0 = d(dvec) 64 -> d(z) 32 -> d(h5) 64)
  lin_bwd(p.dlw, S + OFF_DPRE, S + OFF_Z,  g0, G + P_DLW, G + P_DLB, g0, 64, 32, st);
  lin_bwd(p.elw, S + OFF_ZPRE, S + OFF_H5, g0, G + P_ELW, G + P_ELB, g0, 32, 64, st);
  // encoder conv 6 (enc_w[4]) @4x4, no pool
  for (int i = threadIdx.x; i < 64; i += NTHREADS)
    g1[i] = g0[i] * dswish_(S[OFF_Y5 + i]);
  __syncthreads();
  conv3x3_db(g1, 4, 16, G + P_ENCB + 20);
  conv3x3_dw<4, 4>(g1, S + OFF_H4, 4, 2, G + P_ENCW + 4 * 144);
  __syncthreads();
  conv3x3_dx<4, 4>(g1, p.ew + 4 * 144, 4, 2, g0, wf);  // g0 = d(H4)
  int s2 = 8;
  for (int k = 4; k >= 1; --k) {
    const int wl2 = 7 - k;  // log2(s2)
    k_pool_bwd(g0, S + ENCY[k], g1, 4, s2, wl2);
    __syncthreads();
    conv3x3_db(g1, 4, s2 * s2, G + P_ENCB + k * 4);
    conv3x3_dw<4, 4>(g1, S + ENCH[k - 1], s2, wl2, G + P_ENCW + (k - 1) * 144);
    __syncthreads();
    conv3x3_dx<4, 4>(g1, p.ew + (k - 1) * 144, s2, wl2, g0, wf);  // g0 = d(H_{k-1})
    s2 <<= 1;
  }
  // enc conv0: pool_bwd through Y0, then dW only (input is x)
  k_pool_bwd(g0, S + OFF_Y0, g1, 4, 128, 7);
  __syncthreads();
  conv3x3_db(g1, 4, 16384, G + P_ENCB);
  conv3x3_dw<1, 4>(g1, xs, 128, 7, G + P_ENCW0);
  __syncthreads();
}

__device__ void upd_range(float* dst, const float* src, const float* g, int n) {
  for (int i = threadIdx.x; i < n; i += NTHREADS) dst[i] = src[i] - 0.01f * g[i];
}

// ---------------- one block per sample ----------------
__global__ __launch_bounds__(NTHREADS, 1) void maml_ae_kernel(
    const float* __restrict__ x, const float* __restrict__ ew0,
    const float* __restrict__ ew, const float* __restrict__ eb,
    const float* __restrict__ elw, const float* __restrict__ elb,
    const float* __restrict__ dlw, const float* __restrict__ dlb,
    const float* __restrict__ dw, const float* __restrict__ db,
    const float* __restrict__ dwo, const float* __restrict__ dbo,
    float* __restrict__ out, float* __restrict__ ws) {
  const int s = blockIdx.x;
  float* S = ws + (size_t)s * PER_SAMPLE;
  float* G = S + OFF_GRAD;
  const float* xs = x + (size_t)s * 16384;

  for (int i = threadIdx.x; i < P_TOTAL; i += NTHREADS) G[i] = 0.0f;
  __syncthreads();

  Pm p0{ew0, ew, eb, elw, elb, dlw, dlb, dw, db, dwo, dbo};
  ae_fwd(xs, p0, S, S + OFF_OUT1);  // inner forward (stash activations)
  ae_bwd(xs, p0, S, G);             // inner backward (param grads)
  __syncthreads();

  float* U = S + OFF_UPD;  // p' = p - LR * g  (packed per-sample params)
  upd_range(U + P_ENCW0, ew0, G + P_ENCW0, 36);
  upd_range(U + P_ENCW,  ew,  G + P_ENCW,  720);
  upd_range(U + P_ENCB,  eb,  G + P_ENCB,  24);
  upd_range(U + P_ELW,   elw, G + P_ELW,   2048);
  upd_range(U + P_ELB,   elb, G + P_ELB,   32);
  upd_range(U + P_DLW,   dlw, G + P_DLW,   2048);
  upd_range(U + P_DLB,   dlb, G + P_DLB,   64);
  upd_range(U + P_DECW,  dw,  G + P_DECW,  720);
  upd_range(U + P_DECB,  db,  G + P_DECB,  20);
  upd_range(U + P_DWO,   dwo, G + P_DWO,   36);
  upd_range(U + P_DBO,   dbo, G + P_DBO,   1);
  __syncthreads();

  Pm p1{U + P_ENCW0, U + P_ENCW, U + P_ENCB, U + P_ELW, U + P_ELB, U + P_DLW,
        U + P_DLB,   U + P_DECW, U + P_DECB, U + P_DWO, U + P_DBO};
  ae_fwd(xs, p1, S, out + (size_t)s * 16384);  // post-update forward -> output
}

extern "C" void kernel_launch(void* const* d_in, const int* in_sizes, int n_in,
                              void* d_out, int out_size, void* d_ws, size_t ws_size,
                              hipStream_t stream) {
  (void)in_sizes; (void)n_in; (void)out_size; (void)ws_size;
  maml_ae_kernel<<<128, NTHREADS, 0, stream>>>(
      (const float*)d_in[0], (const float*)d_in[1], (const float*)d_in[2],
      (const float*)d_in[3], (const float*)d_in[4], (const float*)d_in[5],
      (const float*)d_in[6], (const float*)d_in[7], (const float*)d_in[8],
      (const float*)d_in[9], (const float*)d_in[10], (const float*)d_in[11],
      (float*)d_out, (float*)d_ws);
}